// Qwen3DEncoder_33406255629076
// MI455X (gfx1250) — compile-verified
//
#include <hip/hip_runtime.h>

// ---------------- problem constants ----------------
#define NB     8
#define HD_    384
#define WD_    512
#define GH_    48
#define GW_    64
#define CH     2048
#define NPTS   (NB*GH_*GW_)      // 24576
#define NSORT  32768             // next pow2, padded with SENT
#define MAXV   (NPTS+1)          // 24577 voxel slots
#define SENT   0x7FFFFFFFFFFFFFFFLL
#define VOXSZ  0.05f
#define MINPTS 2

typedef __attribute__((ext_vector_type(2))) float v2f;
typedef __attribute__((ext_vector_type(8))) float v8f;
typedef __attribute__((__vector_size__(4 * sizeof(int)))) int v4i;

#if defined(__has_builtin)
#  if __has_builtin(__builtin_amdgcn_global_load_async_to_lds_b128) && \
      __has_builtin(__builtin_amdgcn_s_wait_asynccnt)
#    define USE_ASYNC_LDS 1
#  endif
#endif

// ---------------------------------------------------------------------------
// K1: per-point geometry. Bilinear depth sample, unproject, then the
// extrinsic transform done as a real matrix op on the WMMA unit:
//   D(16x16) = A(16 points x 4 hom) * B(4 x 16 = ext^T padded)
// using V_WMMA_F32_16X16X4_F32, two WMMAs per wave (32 points).
// ---------------------------------------------------------------------------
__global__ __launch_bounds__(256) void k_points(
    const float* __restrict__ depths, const float* __restrict__ intr,
    const float* __restrict__ ext, long long* __restrict__ hout,
    float* __restrict__ centers)
{
  __shared__ float s_hom[256][4];
  __shared__ float s_world[256][3];

  const int tid = threadIdx.x;
  const int p   = blockIdx.x*256 + tid;            // 96 blocks * 256 = 24576
  const int b   = p / (GH_*GW_);                   // whole block shares one b
  const int idx = p % (GH_*GW_);
  const int gy  = idx / GW_, gx = idx % GW_;

  const float px = 8.0f*(float)gx + 4.0f;
  const float py = 8.0f*(float)gy + 4.0f;
  // grid_sample(align_corners=True) coordinates, exactly as the reference
  const float nx = 2.0f*px/(float)WD_ - 1.0f;
  const float ny = 2.0f*py/(float)HD_ - 1.0f;
  const float x  = (nx + 1.0f)*0.5f*(float)(WD_-1);
  const float y  = (ny + 1.0f)*0.5f*(float)(HD_-1);
  const float x0 = floorf(x), y0 = floorf(y);
  const float wxf = x - x0, wyf = y - y0;

  const float* dimg = depths + (size_t)b*HD_*WD_;
  auto gather = [&](float yi, float xi)->float {
    const bool ok = (xi >= 0.0f) && (xi <= (float)(WD_-1)) &&
                    (yi >= 0.0f) && (yi <= (float)(HD_-1));
    const int xc = (int)fminf(fmaxf(xi, 0.0f), (float)(WD_-1));
    const int yc = (int)fminf(fmaxf(yi, 0.0f), (float)(HD_-1));
    return ok ? dimg[yc*WD_ + xc] : 0.0f;
  };
  const float z = gather(y0,      x0     )*(1.0f-wxf)*(1.0f-wyf)
                + gather(y0,      x0+1.0f)*wxf       *(1.0f-wyf)
                + gather(y0+1.0f, x0     )*(1.0f-wxf)*wyf
                + gather(y0+1.0f, x0+1.0f)*wxf       *wyf;

  const float* Kc = intr + b*9;                    // fx=K[0], cx=K[2], fy=K[4], cy=K[5]
  const float camx = (px - Kc[2]) * z / Kc[0];
  const float camy = (py - Kc[5]) * z / Kc[4];

  s_hom[tid][0] = camx; s_hom[tid][1] = camy;
  s_hom[tid][2] = z;    s_hom[tid][3] = 1.0f;
  __syncthreads();

  // ----- WMMA: world_j = sum_k hom_k * ext[j][k]  (B[k][n] = ext[n][k]) -----
  const int lane  = tid & 31;
  const int wbase = tid & ~31;                     // wave's first point in block
  const int col   = lane & 15;                     // N index of this lane
  const float* E  = ext + b*16;

  // B fragment (4x16 f32): VGPR0 = K0 / K2, VGPR1 = K1 / K3 (low/high half-wave)
  v2f bf; bf.x = 0.0f; bf.y = 0.0f;
  if (col < 4) {
    if (lane < 16) { bf.x = E[col*4+0]; bf.y = E[col*4+1]; }
    else           { bf.x = E[col*4+2]; bf.y = E[col*4+3]; }
  }

  for (int g = 0; g < 2; ++g) {                    // two groups of 16 points
    const int mb = wbase + g*16;
    // A fragment (16x4 f32): lanes 0-15 hold K0/K1 of point M=lane,
    // lanes 16-31 hold K2/K3 of point M=lane-16
    v2f af;
    if (lane < 16) { af.x = s_hom[mb+lane][0];    af.y = s_hom[mb+lane][1]; }
    else           { af.x = s_hom[mb+lane-16][2]; af.y = s_hom[mb+lane-16][3]; }
    v8f acc = {0.f,0.f,0.f,0.f,0.f,0.f,0.f,0.f};
    acc = __builtin_amdgcn_wmma_f32_16x16x4_f32(
        /*neg_a=*/false, af, /*neg_b=*/false, bf,
        /*c_mod=*/(short)0, acc, /*reuse_a=*/false, /*reuse_b=*/false);
    // D layout: VGPR r -> M=r (lanes 0-15) / M=8+r (lanes 16-31), N = lane%16
    if (col < 3) {
      const int mo = (lane < 16) ? 0 : 8;
      #pragma unroll
      for (int r = 0; r < 8; ++r) s_world[mb + mo + r][col] = acc[r];
    }
  }
  __syncthreads();

  const float wx3 = s_world[tid][0];
  const float wy3 = s_world[tid][1];
  const float wz3 = s_world[tid][2];

  const long long vx = (long long)floorf(wx3 / VOXSZ);
  const long long vy = (long long)floorf(wy3 / VOXSZ);
  const long long vz = (long long)floorf(wz3 / VOXSZ);
  const bool ok = z > 0.01f;
  hout[p] = ok ? (vx*100000000LL + vy*10000LL + vz) : SENT;
  centers[3*p+0] = ((float)vx + 0.5f)*VOXSZ;
  centers[3*p+1] = ((float)vy + 0.5f)*VOXSZ;
  centers[3*p+2] = ((float)vz + 0.5f)*VOXSZ;
}

// ---------------------------------------------------------------------------
// K2: pad hashes into the sort buffer
// ---------------------------------------------------------------------------
__global__ void k_fill(const long long* __restrict__ h, long long* __restrict__ sb) {
  const int i = blockIdx.x*256 + threadIdx.x;
  if (i < NSORT) sb[i] = (i < NPTS) ? h[i] : SENT;
}

// ---------------------------------------------------------------------------
// K3: one bitonic compare-exchange pass over global memory (launched 120x)
// ---------------------------------------------------------------------------
__global__ void k_bitonic(long long* __restrict__ d, int k, int j) {
  const int i = blockIdx.x*256 + threadIdx.x;
  if (i >= NSORT) return;
  const int l = i ^ j;
  if (l > i) {
    const long long a = d[i], c = d[l];
    const bool up = ((i & k) == 0);
    if ((a > c) == up) { d[i] = c; d[l] = a; }
  }
}

// ---------------------------------------------------------------------------
// K4: flags + single-workgroup scan -> sorted uniques, run starts, n_unique
// (matches jnp.unique's sorted semantics)
// ---------------------------------------------------------------------------
__global__ __launch_bounds__(1024) void k_unique(
    const long long* __restrict__ sb, long long* __restrict__ uniq,
    int* __restrict__ start, int* __restrict__ nuniq)
{
  __shared__ int ps[1024];
  const int t = threadIdx.x;
  const int base = t*32;                            // 1024*32 = 32768
  int local = 0;
  for (int i = 0; i < 32; ++i) {
    const int g = base + i;
    local += ((g == 0) || (sb[g] != sb[g-1])) ? 1 : 0;
  }
  ps[t] = local;
  __syncthreads();
  for (int off = 1; off < 1024; off <<= 1) {        // inclusive scan
    const int v = (t >= off) ? ps[t-off] : 0;
    __syncthreads();
    if (t >= off) ps[t] += v;
    __syncthreads();
  }
  int r = (t == 0) ? 0 : ps[t-1];
  for (int i = 0; i < 32; ++i) {
    const int g = base + i;
    if ((g == 0) || (sb[g] != sb[g-1])) { uniq[r] = sb[g]; start[r] = g; ++r; }
  }
  if (t == 1023) nuniq[0] = ps[1023];
}

// ---------------------------------------------------------------------------
// K5: counts, keep mask (written as float to d_out), SENT-fill unused slots
// ---------------------------------------------------------------------------
__global__ void k_finalize(long long* __restrict__ uniq, const int* __restrict__ start,
                           const int* __restrict__ nuniq, int* __restrict__ cnt,
                           float* __restrict__ keepf)
{
  const int r = blockIdx.x*256 + threadIdx.x;
  if (r >= MAXV) return;
  const int nu = nuniq[0];
  long long u; int c;
  if (r < nu) {
    u = uniq[r];
    const int s = start[r];
    const int e = (r+1 < nu) ? start[r+1] : NSORT;  // SENT slot absorbs padding
    c = e - s;
  } else {
    u = SENT; c = 0; uniq[r] = SENT;                // keep table searchable
  }
  cnt[r] = c;
  keepf[r] = (c >= MINPTS && u != SENT) ? 1.0f : 0.0f;
}

// ---------------------------------------------------------------------------
// K6: zero vp+vf output regions (harness poisons d_out with 0xAA)
// ---------------------------------------------------------------------------
__global__ void k_zero(float* __restrict__ o, long long n) {
  long long i = (long long)blockIdx.x*blockDim.x + threadIdx.x;
  const long long stride = (long long)gridDim.x*blockDim.x;
  for (; i < n; i += stride) o[i] = 0.0f;
}

// ---------------------------------------------------------------------------
// K7: per-point feature aggregation (the bandwidth-dominant step, ~400 MB).
// One block per point: stage the 8 KB feature row into LDS with the CDNA5
// async-to-LDS path, then atomic-add row*(1/cnt) into the voxel row.
// Rows whose voxel is not kept (cnt<2 or SENT) are skipped -> stay zero,
// matching the reference's final masking.
// ---------------------------------------------------------------------------
__global__ __launch_bounds__(256) void k_accum(
    const float* __restrict__ feats, const float* __restrict__ centers,
    const long long* __restrict__ h, const long long* __restrict__ uniq,
    const int* __restrict__ cnt, float* __restrict__ vp, float* __restrict__ vf)
{
  __shared__ __align__(16) float row[CH];
  __shared__ int   s_inv;
  __shared__ float s_scale;
  const int p   = blockIdx.x;
  const int tid = threadIdx.x;

  if (tid == 0) {
    int inv = -1; float sc = 0.0f;
    const long long hp = h[p];
    if (hp != SENT) {
      int lo = 0, hi = MAXV;                        // lower_bound over sorted uniq
      while (lo < hi) { const int mid = (lo+hi)>>1; if (uniq[mid] < hp) lo = mid+1; else hi = mid; }
      if (lo < MAXV && uniq[lo] == hp) {
        const int c = cnt[lo];
        if (c >= MINPTS) { inv = lo; sc = 1.0f/(float)c; }
      }
    }
    s_inv = inv; s_scale = sc;
  }
  __syncthreads();
  const int inv = s_inv;
  if (inv < 0) return;                              // uniform per block
  const float sc = s_scale;
  const float* src = feats + (size_t)p*CH;

#if defined(USE_ASYNC_LDS)
  #pragma unroll
  for (int i = 0; i < 2; ++i) {                     // 2 x b128 = 8 floats/lane
    const int e = tid*8 + i*4;
    __builtin_amdgcn_global_load_async_to_lds_b128(
        (__attribute__((address_space(1))) v4i*)(src + e),
        (__attribute__((address_space(3))) v4i*)(&row[e]),
        0, 0);
  }
  __builtin_amdgcn_s_wait_asynccnt(0);
  __syncthreads();
#else
  for (int i = tid; i < CH/4; i += 256)
    ((float4*)row)[i] = ((const float4*)src)[i];
  __syncthreads();
#endif

  float* dst = vf + (size_t)inv*CH;
  for (int c = tid; c < CH; c += 256)
    unsafeAtomicAdd(&dst[c], row[c]*sc);            // global_atomic_add_f32
  if (tid < 3)
    unsafeAtomicAdd(&vp[inv*3+tid], centers[p*3+tid]*sc);
}

// ---------------------------------------------------------------------------
extern "C" void kernel_launch(void* const* d_in, const int* in_sizes, int n_in,
                              void* d_out, int out_size, void* d_ws, size_t ws_size,
                              hipStream_t stream) {
  const float* features = (const float*)d_in[0];   // (B, 3072, 2048)
  const float* depths   = (const float*)d_in[1];   // (B, 384, 512)
  const float* intr     = (const float*)d_in[2];   // (B, 3, 3)
  const float* ext      = (const float*)d_in[3];   // (B, 4, 4)

  // ---- workspace carve (~1.2 MB) ----
  long long* h     = (long long*)d_ws;             // NPTS
  long long* sb    = h + NPTS;                     // NSORT
  long long* uniq  = sb + NSORT;                   // MAXV
  int*  start      = (int*)(uniq + MAXV);          // MAXV
  int*  cnt        = start + MAXV;                 // MAXV
  int*  nuniq      = cnt + MAXV;                   // 1
  float* centers   = (float*)(nuniq + 1);          // NPTS*3

  // ---- output carve: vp | vf | keep ----
  float* out   = (float*)d_out;
  float* vp    = out;
  float* vf    = out + (size_t)MAXV*3;
  float* keepf = out + (size_t)MAXV*(3+CH);

  k_points<<<NPTS/256, 256, 0, stream>>>(depths, intr, ext, h, centers);
  k_fill  <<<NSORT/256, 256, 0, stream>>>(h, sb);
  for (int k = 2; k <= NSORT; k <<= 1)
    for (int j = k >> 1; j > 0; j >>= 1)
      k_bitonic<<<NSORT/256, 256, 0, stream>>>(sb, k, j);
  k_unique  <<<1, 1024, 0, stream>>>(sb, uniq, start, nuniq);
  k_finalize<<<(MAXV+255)/256, 256, 0, stream>>>(uniq, start, nuniq, cnt, keepf);
  k_zero    <<<2048, 256, 0, stream>>>(out, (long long)MAXV*(3+CH));
  k_accum   <<<NPTS, 256, 0, stream>>>(features, centers, h, uniq, cnt, vp, vf);
}